// GCNModel_56942676410567
// MI455X (gfx1250) — compile-verified
//
#include <hip/hip_runtime.h>
#include <math.h>

typedef __attribute__((ext_vector_type(2))) float v2f;
typedef __attribute__((ext_vector_type(8))) float v8f;

static constexpr int FEAT = 128;   // IN_FEATS == HIDDEN == 128
static constexpr int APAD = 132;   // padded LDS row stride (dwords) to spread banks
static constexpr int ROWS_PER_BLK = 64;

// ---------------------------------------------------------------- utilities
__global__ __launch_bounds__(256) void k_zero(float* __restrict__ p, int n) {
    int i = blockIdx.x * 256 + threadIdx.x;
    if (i < n) p[i] = 0.0f;
}

// degree accumulation: deg_out[src[e]] += 1, deg_in[dst[e]] += 1
__global__ __launch_bounds__(256) void k_degree(const int* __restrict__ src,
                                                const int* __restrict__ dst,
                                                float* __restrict__ degO,
                                                float* __restrict__ degI, int ne) {
    int e = blockIdx.x * 256 + threadIdx.x;
    if (e < ne) {
        atomicAdd(degO + src[e], 1.0f);
        atomicAdd(degI + dst[e], 1.0f);
    }
}

// in-place: deg -> 1/sqrt(max(deg,1))
__global__ __launch_bounds__(256) void k_rsqrt_clip(float* __restrict__ degO,
                                                    float* __restrict__ degI, int n) {
    int i = blockIdx.x * 256 + threadIdx.x;
    if (i < n) {
        float a = degO[i]; a = a < 1.0f ? 1.0f : a;
        degO[i] = 1.0f / sqrtf(a);
        float b = degI[i]; b = b < 1.0f ? 1.0f : b;
        degI[i] = 1.0f / sqrtf(b);
    }
}

// ---------------------------------------------------------------- GEMM (WMMA fp32)
// Y[row, :] = (X[row, :] * rs[row]) @ W       X:[n,128] W:[128,128] Y:[n,128]
// Block: 256 threads = 8 waves covering 64 rows; wave w owns N-tile [16w,16w+16).
// K-outer loop: each K-step loads one B fragment (shared by 4 M-tiles) and issues
// 4 independent V_WMMA_F32_16X16X4_F32 (no accumulator RAW between consecutive WMMAs).
__global__ __launch_bounds__(256) void k_gemm_scaled(const float* __restrict__ X,
                                                     const float* __restrict__ rs,
                                                     const float* __restrict__ W,
                                                     float* __restrict__ Y, int nrows) {
    __shared__ float Alds[ROWS_PER_BLK * APAD];   // 64*132*4 = 33,792 B

    const int rowbase = blockIdx.x * ROWS_PER_BLK;
    const int t = threadIdx.x;

    // stage A tile (scaled) into LDS, float4 granularity: 64 rows * 32 float4
    for (int i = t; i < ROWS_PER_BLK * (FEAT / 4); i += 256) {
        int r  = i >> 5;               // local row
        int c4 = (i & 31) << 2;        // col (multiple of 4)
        int row = rowbase + r;
        float4 v = {0.f, 0.f, 0.f, 0.f};
        float s = 0.f;
        if (row < nrows) {
            v = *(const float4*)(X + (size_t)row * FEAT + c4);
            s = rs[row];
        }
        float* dstp = Alds + r * APAD + c4;
        dstp[0] = v.x * s; dstp[1] = v.y * s; dstp[2] = v.z * s; dstp[3] = v.w * s;
    }
    __syncthreads();

    const int wave = t >> 5;           // 0..7 -> N tile
    const int lane = t & 31;
    const int half = lane >> 4;        // 0: K=k,k+1   1: K=k+2,k+3
    const int l16  = lane & 15;
    const int ncol = wave * 16 + l16;  // global output column for B/D

    v8f acc0 = {}, acc1 = {}, acc2 = {}, acc3 = {};
    const float* a0 = Alds + (0 * 16 + l16) * APAD;
    const float* a1 = Alds + (1 * 16 + l16) * APAD;
    const float* a2 = Alds + (2 * 16 + l16) * APAD;
    const float* a3 = Alds + (3 * 16 + l16) * APAD;

    #pragma unroll 4
    for (int k = 0; k < FEAT; k += 4) {
        const int ka = k + half * 2;
        v2f b;                          // B fragment shared by all 4 M-tiles
        b.x = W[ka * FEAT + ncol];      // W row-major [k][n]
        b.y = W[(ka + 1) * FEAT + ncol];
        v2f a;
        a.x = a0[ka]; a.y = a0[ka + 1];
        acc0 = __builtin_amdgcn_wmma_f32_16x16x4_f32(false, a, false, b, (short)0, acc0, false, false);
        a.x = a1[ka]; a.y = a1[ka + 1];
        acc1 = __builtin_amdgcn_wmma_f32_16x16x4_f32(false, a, false, b, (short)0, acc1, false, false);
        a.x = a2[ka]; a.y = a2[ka + 1];
        acc2 = __builtin_amdgcn_wmma_f32_16x16x4_f32(false, a, false, b, (short)0, acc2, false, false);
        a.x = a3[ka]; a.y = a3[ka + 1];
        acc3 = __builtin_amdgcn_wmma_f32_16x16x4_f32(false, a, false, b, (short)0, acc3, false, false);
    }

    // D layout: VGPR r -> M = r + 8*half, N = l16
    const int mrow0 = rowbase + half * 8;
    if (rowbase + ROWS_PER_BLK <= nrows) {
        // full tile: unguarded, uniform path
        #pragma unroll
        for (int r = 0; r < 8; ++r) {
            Y[(size_t)(mrow0 +  0 + r) * FEAT + ncol] = acc0[r];
            Y[(size_t)(mrow0 + 16 + r) * FEAT + ncol] = acc1[r];
            Y[(size_t)(mrow0 + 32 + r) * FEAT + ncol] = acc2[r];
            Y[(size_t)(mrow0 + 48 + r) * FEAT + ncol] = acc3[r];
        }
    } else {
        #pragma unroll
        for (int r = 0; r < 8; ++r) {
            int r0 = mrow0 + r;
            if (r0 +  0 < nrows) Y[(size_t)(r0 +  0) * FEAT + ncol] = acc0[r];
            if (r0 + 16 < nrows) Y[(size_t)(r0 + 16) * FEAT + ncol] = acc1[r];
            if (r0 + 32 < nrows) Y[(size_t)(r0 + 32) * FEAT + ncol] = acc2[r];
            if (r0 + 48 < nrows) Y[(size_t)(r0 + 48) * FEAT + ncol] = acc3[r];
        }
    }
}

// ---------------------------------------------------------------- edge scatter-add
// agg[dst[e], :] += H[src[e], :]   one wave per edge, lane handles 4 floats
__global__ __launch_bounds__(256) void k_scatter(const float* __restrict__ H,
                                                 const int* __restrict__ src,
                                                 const int* __restrict__ dst,
                                                 float* __restrict__ agg, int ne) {
    int e = blockIdx.x * 8 + (threadIdx.x >> 5);
    if (e >= ne) return;
    int lane = threadIdx.x & 31;
    int s = src[e], d = dst[e];
    float4 v = *(const float4*)(H + (size_t)s * FEAT + lane * 4);
    float* p = agg + (size_t)d * FEAT + lane * 4;
    atomicAdd(p + 0, v.x);
    atomicAdd(p + 1, v.y);
    atomicAdd(p + 2, v.z);
    atomicAdd(p + 3, v.w);
}

// out = relu(agg * rs_in[row] + bias[col])
__global__ __launch_bounds__(256) void k_finalize(const float* __restrict__ agg,
                                                  const float* __restrict__ rsI,
                                                  const float* __restrict__ bias,
                                                  float* __restrict__ out, int n) {
    int i = blockIdx.x * 256 + threadIdx.x;
    if (i < n) {
        int row = i >> 7;        // /128
        int col = i & 127;
        float v = agg[i] * rsI[row] + bias[col];
        out[i] = v > 0.f ? v : 0.f;
    }
}

// logits = H @ Wc + bc   Wc:[128,4]  one thread per (node, class)
__global__ __launch_bounds__(256) void k_classify(const float* __restrict__ H,
                                                  const float* __restrict__ Wc,
                                                  const float* __restrict__ bc,
                                                  float* __restrict__ out, int n4) {
    int i = blockIdx.x * 256 + threadIdx.x;
    if (i >= n4) return;
    int node = i >> 2;
    int c = i & 3;
    const float* h = H + (size_t)node * FEAT;
    float sum = 0.f;
    #pragma unroll 8
    for (int k = 0; k < FEAT; ++k) sum += h[k] * Wc[k * 4 + c];
    out[i] = sum + bc[c];
}

// ---------------------------------------------------------------- launch
extern "C" void kernel_launch(void* const* d_in, const int* in_sizes, int n_in,
                              void* d_out, int out_size, void* d_ws, size_t ws_size,
                              hipStream_t stream) {
    const float* feat = (const float*)d_in[0];
    const int*   src  = (const int*)d_in[1];
    const int*   dst  = (const int*)d_in[2];
    const float* W1   = (const float*)d_in[3];
    const float* b1   = (const float*)d_in[4];
    const float* W2   = (const float*)d_in[5];
    const float* b2   = (const float*)d_in[6];
    const float* Wc   = (const float*)d_in[7];
    const float* bc   = (const float*)d_in[8];
    float* out = (float*)d_out;

    const int nnodes = in_sizes[0] / FEAT;
    const int nedges = in_sizes[1];
    const size_t nAlign = ((size_t)nnodes + 3) & ~(size_t)3;
    const size_t nfeat = (size_t)nnodes * FEAT;

    float* rsO  = (float*)d_ws;        // deg_out -> rsqrt
    float* rsI  = rsO + nAlign;        // deg_in  -> rsqrt
    float* bufA = rsI + nAlign;        // [n,128]
    float* bufB = bufA + nfeat;        // [n,128]

    const int B = 256;
    const int gN      = (int)((nfeat + B - 1) / B);
    const int gNode   = (nnodes + B - 1) / B;
    const int gEdgeT  = (nedges + B - 1) / B;
    const int gEdgeW  = (nedges + 7) / 8;        // 1 wave per edge
    const int gGemm   = (nnodes + ROWS_PER_BLK - 1) / ROWS_PER_BLK;
    const int gCls    = (nnodes * 4 + B - 1) / B;

    // degrees -> 1/sqrt(clip(deg,1))
    k_zero<<<(int)((2 * nAlign + B - 1) / B), B, 0, stream>>>(rsO, (int)(2 * nAlign));
    k_degree<<<gEdgeT, B, 0, stream>>>(src, dst, rsO, rsI, nedges);
    k_rsqrt_clip<<<gNode, B, 0, stream>>>(rsO, rsI, nnodes);

    // ---- layer 1
    k_gemm_scaled<<<gGemm, B, 0, stream>>>(feat, rsO, W1, bufA, nnodes);
    k_zero<<<gN, B, 0, stream>>>(bufB, (int)nfeat);
    k_scatter<<<gEdgeW, B, 0, stream>>>(bufA, src, dst, bufB, nedges);
    k_finalize<<<gN, B, 0, stream>>>(bufB, rsI, b1, bufA, (int)nfeat);   // bufA = h1

    // ---- layer 2
    k_gemm_scaled<<<gGemm, B, 0, stream>>>(bufA, rsO, W2, bufB, nnodes);
    k_zero<<<gN, B, 0, stream>>>(bufA, (int)nfeat);
    k_scatter<<<gEdgeW, B, 0, stream>>>(bufB, src, dst, bufA, nedges);
    k_finalize<<<gN, B, 0, stream>>>(bufA, rsI, b2, bufB, (int)nfeat);   // bufB = h2

    // ---- classifier
    k_classify<<<gCls, B, 0, stream>>>(bufB, Wc, bc, out, nnodes * 4);
}